// TemporalPyramidPoolingDet_67284957659780
// MI455X (gfx1250) — compile-verified
//
#include <hip/hip_runtime.h>
#include <math.h>

#define B_    16
#define D_    512
#define T_    4096
#define H_    1024
#define DOUT_ 512
#define NBINS_ 31
#define MROWS_ (B_ * NBINS_)   // 496

typedef __attribute__((ext_vector_type(2))) float v2f;
typedef __attribute__((ext_vector_type(8))) float v8f;

// ---------------------------------------------------------------- zero scratch
__global__ void zero_kernel(float* p, int n) {
  int i = blockIdx.x * blockDim.x + threadIdx.x;
  if (i < n) p[i] = 0.f;
}

// ------------------------------------------------- per-batch Tv, fine bounds, cnt
// Fine (level-16) boundaries fb[0..16]; all coarser-level boundaries coincide
// exactly (power-of-2 exact f32 arithmetic), so coarse bins are sums of fine bins.
__global__ void prep_kernel(const unsigned char* __restrict__ mask,
                            float* __restrict__ Tv, int* __restrict__ fb,
                            float* __restrict__ cnt) {
  __shared__ float red[256];
  __shared__ int   sfb[17];
  __shared__ float scnt[16];
  int b = blockIdx.x, tid = threadIdx.x;
  float local = 0.f;
  for (int t = tid; t < T_; t += 256)
    local += (mask[b * T_ + t] == 0) ? 1.f : 0.f;   // w = ~time_mask
  red[tid] = local; __syncthreads();
  for (int s = 128; s > 0; s >>= 1) { if (tid < s) red[tid] += red[tid + s]; __syncthreads(); }
  if (tid == 0) {
    float Tvv = roundf(red[0]);
    Tv[b] = Tvv;
    float scale = (float)T_ / fmaxf(Tvv, 1.f);
    for (int k = 0; k <= 16; ++k) {
      float idxk = floorf(Tvv * (float)k / 16.f);
      float s2 = floorf(idxk * scale);
      int v = (int)fminf(fmaxf(s2, 0.f), (float)T_);
      sfb[k] = v; fb[b * 17 + k] = v;
    }
  }
  if (tid < 16) scnt[tid] = 0.f;
  __syncthreads();
  for (int t = tid; t < T_; t += 256) {
    if (mask[b * T_ + t] == 0) {
      for (int k = 0; k < 16; ++k) {
        if (t < sfb[k + 1]) { if (t >= sfb[k]) atomicAdd(&scnt[k], 1.f); break; }
      }
    }
  }
  __syncthreads();
  if (tid < 16) cnt[b * 16 + tid] = scnt[tid];
}

// --------------------------------------------- fine bin sums (single pass over z)
// grid (B, T/256, 2); each thread owns one d-column over a 256-long t-chunk,
// streams float4 loads, flushes per-bin accumulators with global f32 atomics.
__global__ void binsum_kernel(const float* __restrict__ z,
                              const unsigned char* __restrict__ mask,
                              const int* __restrict__ fb,
                              float* __restrict__ fineS, float* __restrict__ fineS2) {
  __shared__ int sfb[17];
  int b  = blockIdx.x;
  int t0 = blockIdx.y * 256;
  int d  = blockIdx.z * 256 + threadIdx.x;
  if (threadIdx.x < 17) sfb[threadIdx.x] = fb[b * 17 + threadIdx.x];
  __syncthreads();
  const float* zrow = z + ((size_t)b * D_ + d) * T_;
  const unsigned char* mrow = mask + (size_t)b * T_;
  int bin = 16;                                    // 16 == past last boundary
  for (int k = 0; k < 16; ++k) if (t0 < sfb[k + 1]) { bin = k; break; }
  float az = 0.f, az2 = 0.f;
  for (int t = t0; t < t0 + 256; t += 4) {
    float4 zv = *(const float4*)(zrow + t);
    unsigned int mw = *(const unsigned int*)(mrow + t);
    float ze[4] = {zv.x, zv.y, zv.z, zv.w};
#pragma unroll
    for (int e = 0; e < 4; ++e) {
      int tt = t + e;
      while (bin < 16 && tt >= sfb[bin + 1]) {
        if (az != 0.f || az2 != 0.f) {
          atomicAdd(&fineS [((size_t)b * 16 + bin) * D_ + d], az);
          atomicAdd(&fineS2[((size_t)b * 16 + bin) * D_ + d], az2);
          az = az2 = 0.f;
        }
        ++bin;
      }
      if (bin < 16 && ((mw >> (8 * e)) & 0xffu) == 0u) {
        float zz = ze[e];
        az += zz; az2 += zz * zz;
      }
    }
  }
  if (bin < 16 && (az != 0.f || az2 != 0.f)) {
    atomicAdd(&fineS [((size_t)b * 16 + bin) * D_ + d], az);
    atomicAdd(&fineS2[((size_t)b * 16 + bin) * D_ + d], az2);
  }
}

// ----------------------------------------------------- build feat (mean || std)
__global__ void feat_kernel(const float* __restrict__ fineS, const float* __restrict__ fineS2,
                            const float* __restrict__ cnt, const int* __restrict__ fb,
                            const float* __restrict__ Tv, float* __restrict__ feat) {
  int b = blockIdx.x, bi = blockIdx.y, tid = threadIdx.x;
  int nb, j;
  if      (bi < 1)  { nb = 1;  j = bi;      }
  else if (bi < 3)  { nb = 2;  j = bi - 1;  }
  else if (bi < 7)  { nb = 4;  j = bi - 3;  }
  else if (bi < 15) { nb = 8;  j = bi - 7;  }
  else              { nb = 16; j = bi - 15; }
  int r = 16 / nb;
  int f0 = j * r, f1 = f0 + r;
  int s_idx = fb[b * 17 + f0], e_idx = fb[b * 17 + f1];
  float Tvv = Tv[b];
  float c = 0.f;
  for (int k = f0; k < f1; ++k) c += cnt[b * 16 + k];
  float cc = fmaxf(c, 1.f);
  bool valid = (e_idx > s_idx) && (Tvv > 0.f);
  for (int d = tid; d < D_; d += 256) {
    float sz = 0.f, sz2 = 0.f;
    for (int k = f0; k < f1; ++k) {
      sz  += fineS [((size_t)b * 16 + k) * D_ + d];
      sz2 += fineS2[((size_t)b * 16 + k) * D_ + d];
    }
    float mu  = sz / cc;
    float var = sz2 / cc - mu * mu;
    float sd  = sqrtf(fmaxf(var, 0.f) + 1e-6f);
    size_t rowoff = ((size_t)(b * NBINS_ + bi)) * H_;
    feat[rowoff + d]       = valid ? mu : 0.f;
    feat[rowoff + D_ + d]  = valid ? sd : 0.f;
  }
}

// --------------------------------------- f32 WMMA GEMM: C = epi(A @ B + bias)
// One wave per 16x16 tile; K-loop of V_WMMA_F32_16X16X4_F32.
// A frag (16x4): lane<16 -> {A[r][k0], A[r][k0+1]}, lane>=16 -> {A[r][k0+2], A[r][k0+3]}
// B frag (4x16): lane<16 -> {B[k0][c], B[k0+1][c]}, lane>=16 -> {B[k0+2][c], B[k0+3][c]}
// C frag: vgpr v, lane<16 -> row v, lane>=16 -> row v+8; col = lane&15.
__global__ void gemm_wmma_kernel(const float* __restrict__ A, const float* __restrict__ Bm,
                                 const float* __restrict__ bias, float* __restrict__ C,
                                 int M, int N, int K, int epi) {
  int wave = threadIdx.x >> 5;
  int lane = threadIdx.x & 31;
  int tile = blockIdx.x * 8 + wave;
  int tilesN = N >> 4;
  int tm = (tile / tilesN) << 4;
  int tn = (tile % tilesN) << 4;
  int col  = lane & 15;            // also A row index (lane&15)
  int koff = (lane >> 4) << 1;
  const float* aptr = A  + (size_t)(tm + col) * K + koff;
  const float* bptr = Bm + (size_t)koff * N + tn + col;
  v8f c = {0.f, 0.f, 0.f, 0.f, 0.f, 0.f, 0.f, 0.f};
  for (int k0 = 0; k0 < K; k0 += 4) {
    v2f a = *(const v2f*)aptr;           // 8B aligned: koff in {0,2}
    v2f bb; bb.x = bptr[0]; bb.y = bptr[N];
    c = __builtin_amdgcn_wmma_f32_16x16x4_f32(false, a, false, bb,
                                              (short)0, c, false, false);
    aptr += 4;
    bptr += (size_t)4 * N;
  }
  float bv = bias[tn + col];
  int rbase = tm + ((lane >> 4) << 3);
#pragma unroll
  for (int v = 0; v < 8; ++v) {
    float x = c[v] + bv;
    if (epi == 0) {            // exact GELU: x * 0.5 * (1 + erf(x/sqrt(2)))
      x = 0.5f * x * (1.f + erff(x * 0.70710678118654752440f));
    } else {                   // nan_to_num
      if (isnan(x)) x = 0.f;
      else if (isinf(x)) x = (x > 0.f) ? 3.4028234663852886e38f : -3.4028234663852886e38f;
    }
    C[(size_t)(rbase + v) * N + tn + col] = x;
  }
}

// ------------------------------------------------------------------- launcher
extern "C" void kernel_launch(void* const* d_in, const int* in_sizes, int n_in,
                              void* d_out, int out_size, void* d_ws, size_t ws_size,
                              hipStream_t stream) {
  const float*         z    = (const float*)d_in[0];
  const unsigned char* mask = (const unsigned char*)d_in[1];  // jnp bool -> 1 byte
  const float*         W1   = (const float*)d_in[2];
  const float*         b1   = (const float*)d_in[3];
  const float*         W2   = (const float*)d_in[4];
  const float*         b2   = (const float*)d_in[5];
  float* out = (float*)d_out;
  char*  ws  = (char*)d_ws;

  float* Tv     = (float*)(ws + 0);        // 16 f
  int*   fb     = (int*)  (ws + 256);      // 16*17 i32
  float* cnt    = (float*)(ws + 2048);     // 16*16 f
  float* fineS  = (float*)(ws + 4096);     // 16*16*512 f
  float* fineS2 = fineS  + (size_t)B_ * 16 * D_;
  float* feat   = fineS2 + (size_t)B_ * 16 * D_;   // 496*1024 f
  float* h      = feat   + (size_t)MROWS_ * H_;    // 496*1024 f

  int nzero = 2 * B_ * 16 * D_;
  zero_kernel<<<(nzero + 255) / 256, 256, 0, stream>>>(fineS, nzero);
  prep_kernel<<<B_, 256, 0, stream>>>(mask, Tv, fb, cnt);
  binsum_kernel<<<dim3(B_, T_ / 256, 2), 256, 0, stream>>>(z, mask, fb, fineS, fineS2);
  feat_kernel<<<dim3(B_, NBINS_), 256, 0, stream>>>(fineS, fineS2, cnt, fb, Tv, feat);
  // GEMM1: 496x1024x1024, fused bias + exact GELU -> h
  gemm_wmma_kernel<<<(MROWS_ / 16) * (H_ / 16) / 8, 256, 0, stream>>>(
      feat, W1, b1, h, MROWS_, H_, H_, 0);
  // GEMM2: 496x512x1024, fused bias + nan_to_num -> out
  gemm_wmma_kernel<<<(MROWS_ / 16) * (DOUT_ / 16) / 8, 256, 0, stream>>>(
      h, W2, b2, out, MROWS_, DOUT_, H_, 1);
}